// MultiGridTokenPooler_30648886624913
// MI455X (gfx1250) — compile-verified
//
#include <hip/hip_runtime.h>
#include <math.h>

typedef __attribute__((ext_vector_type(2))) float v2f;
typedef __attribute__((ext_vector_type(8))) float v8f;

// Grid levels: 16^3, 32^3, 64^3
#define NB0 4096
#define NB1 32768
#define NB2 262144
#define NBTOT 299008           // NB0+NB1+NB2
#define C 128
#define S 16                   // queries per level
#define SROWS 48               // 3 levels * 16
// workspace layout (float offsets)
#define MINMAX_OFF 0           // 6 u32 encoded min/max
#define COUNTS_OFF 16          // NBTOT floats
#define STATS_OFF  (16 + NBTOT)            // 48*2 floats (rowmax, rowsum)
#define SUMS_OFF   (16 + NBTOT + 96 + 16)  // = 299136, 16B aligned; NBTOT*128 floats
#define LOGITS_OFF (SUMS_OFF + (size_t)NBTOT * 128)   // 16*NBTOT floats

__device__ __forceinline__ unsigned encf(float f) {
  unsigned u = __float_as_uint(f);
  return (u & 0x80000000u) ? ~u : (u | 0x80000000u);
}
__device__ __forceinline__ float decf(unsigned u) {
  return (u & 0x80000000u) ? __uint_as_float(u ^ 0x80000000u) : __uint_as_float(~u);
}

__device__ __forceinline__ void lvl_params(int lvl, int& binoff, int& V, int& dim) {
  if (lvl == 0)      { binoff = 0;        V = NB0; dim = 16; }
  else if (lvl == 1) { binoff = NB0;      V = NB1; dim = 32; }
  else               { binoff = NB0+NB1;  V = NB2; dim = 64; }
}

// ---------------- init: encoded min/max seeds + zero this batch's output ----
__global__ void k_init(unsigned* __restrict__ mm, float* __restrict__ outb) {
  int t = threadIdx.x;
  if (t < 3)       mm[t] = 0xFFFFFFFFu;  // min slots
  else if (t < 6)  mm[t] = 0u;           // max slots
  for (int i = t; i < SROWS * C; i += 256) outb[i] = 0.f;
}

// ---------------- per-batch coordinate min/max --------------------------
__global__ void k_minmax(const float* __restrict__ coords, int N,
                         unsigned* __restrict__ mm) {
  int gid = blockIdx.x * blockDim.x + threadIdx.x;
  int stride = gridDim.x * blockDim.x;
  float mn[3] = { __builtin_inff(),  __builtin_inff(),  __builtin_inff() };
  float mx[3] = { -__builtin_inff(), -__builtin_inff(), -__builtin_inff() };
  for (int p = gid; p < N; p += stride) {
    #pragma unroll
    for (int a = 0; a < 3; ++a) {
      float v = coords[(size_t)p * 3 + a];
      mn[a] = fminf(mn[a], v);
      mx[a] = fmaxf(mx[a], v);
    }
  }
  #pragma unroll
  for (int a = 0; a < 3; ++a) {
    atomicMin(&mm[a],     encf(mn[a]));
    atomicMax(&mm[3 + a], encf(mx[a]));
  }
}

// ------- one pass over tokens, scatter into all three levels' sums ------
__global__ void k_bin(const float* __restrict__ coords,
                      const float* __restrict__ tokens, int N,
                      const unsigned* __restrict__ mm, float* __restrict__ ws) {
  int wave = threadIdx.x >> 5, lane = threadIdx.x & 31;
  int p = blockIdx.x * 8 + wave;
  if (p >= N) return;  // wave-uniform

  float mn[3], dn[3];
  #pragma unroll
  for (int a = 0; a < 3; ++a) {
    mn[a] = decf(mm[a]);
    dn[a] = fmaxf(decf(mm[3 + a]) - mn[a], 1e-6f);
  }
  float nc[3];
  #pragma unroll
  for (int a = 0; a < 3; ++a)
    nc[a] = (coords[(size_t)p * 3 + a] - mn[a]) / dn[a];

  int bins[3];
  #pragma unroll
  for (int lvl = 0; lvl < 3; ++lvl) {
    int binoff, V, d; lvl_params(lvl, binoff, V, d);
    float fd = (float)d;
    int ix = (int)fminf(fmaxf(floorf(nc[0] * fd), 0.f), fd - 1.f);
    int iy = (int)fminf(fmaxf(floorf(nc[1] * fd), 0.f), fd - 1.f);
    int iz = (int)fminf(fmaxf(floorf(nc[2] * fd), 0.f), fd - 1.f);
    bins[lvl] = binoff + (ix * d + iy) * d + iz;
  }

  const float4 t = ((const float4*)(tokens + (size_t)p * C))[lane];
  float* sums = ws + SUMS_OFF;
  #pragma unroll
  for (int lvl = 0; lvl < 3; ++lvl) {
    size_t base = (size_t)bins[lvl] * C + lane * 4;
    atomicAdd(&sums[base + 0], t.x);
    atomicAdd(&sums[base + 1], t.y);
    atomicAdd(&sums[base + 2], t.z);
    atomicAdd(&sums[base + 3], t.w);
  }
  if (lane == 0) {
    #pragma unroll
    for (int lvl = 0; lvl < 3; ++lvl)
      atomicAdd(&ws[COUNTS_OFF + bins[lvl]], 1.f);
  }
}

// ---------------- sums -> means (in place) ------------------------------
__global__ void k_mean(float* __restrict__ ws) {
  size_t idx = (size_t)blockIdx.x * 256 + threadIdx.x;  // bin*32 + float4 slot
  int bin = (int)(idx >> 5);
  if (bin >= NBTOT) return;
  float inv = 1.f / fmaxf(ws[COUNTS_OFF + bin], 1.f);
  float4* s = (float4*)(ws + SUMS_OFF) + idx;
  float4 v = *s;
  v.x *= inv; v.y *= inv; v.z *= inv; v.w *= inv;
  *s = v;
}

// -------- logits = (Q @ mean^T) * scale via V_WMMA_F32_16X16X4_F32 ------
// one wave per 16-bin tile; A = Q (16x128), B = mean^T tile, K loop of 4
__global__ void k_logits(const float* __restrict__ q0,
                         const float* __restrict__ q1,
                         const float* __restrict__ q2,
                         float* __restrict__ ws) {
  int wid = blockIdx.x * 8 + (threadIdx.x >> 5);
  int lane = threadIdx.x & 31;
  int lvl, tile;
  if (wid < 256)        { lvl = 0; tile = wid; }
  else if (wid < 2304)  { lvl = 1; tile = wid - 256; }
  else                  { lvl = 2; tile = wid - 2304; }
  int binoff, V, d; lvl_params(lvl, binoff, V, d);
  const float* q = (lvl == 0) ? q0 : (lvl == 1 ? q1 : q2);
  const float* means = ws + SUMS_OFF;

  int n = lane & 15;                  // A row / B column for this lane slot
  int khi = (lane >> 4) << 1;         // lanes 16-31 hold K+2,K+3

  const float* ap = q + (size_t)n * C + khi;
  const float* bp = means + (size_t)(binoff + tile * 16 + n) * C + khi;

  v8f acc = {};
  #pragma unroll 8
  for (int it = 0; it < C / 4; ++it) {
    v2f a = *(const v2f*)ap;
    v2f b = *(const v2f*)bp;
    acc = __builtin_amdgcn_wmma_f32_16x16x4_f32(false, a, false, b,
                                                (short)0, acc, false, false);
    ap += 4; bp += 4;
  }

  float cnt = ws[COUNTS_OFF + binoff + tile * 16 + n];
  const float scale = 0.08838834764831845f;   // 128^-0.5
  float* lg = ws + LOGITS_OFF + (size_t)S * binoff;
  int mbase = (lane >> 4) << 3;
  #pragma unroll
  for (int r = 0; r < 8; ++r) {
    float v = acc[r] * scale;
    if (cnt <= 0.f) v = -__builtin_inff();
    lg[(size_t)(r + mbase) * V + tile * 16 + n] = v;
  }
}

// ---------------- per (level,row) softmax max + sumexp ------------------
__global__ void k_stats(float* __restrict__ ws) {
  __shared__ float red[256];
  int bid = blockIdx.x, t = threadIdx.x;
  int lvl = bid >> 4, s = bid & 15;
  int binoff, V, d; lvl_params(lvl, binoff, V, d);
  const float* row = ws + LOGITS_OFF + (size_t)S * binoff + (size_t)s * V;

  float m = -__builtin_inff();
  for (int v = t; v < V; v += 256) m = fmaxf(m, row[v]);
  red[t] = m; __syncthreads();
  for (int o = 128; o > 0; o >>= 1) {
    if (t < o) red[t] = fmaxf(red[t], red[t + o]);
    __syncthreads();
  }
  float rowmax = red[0]; __syncthreads();

  float sum = 0.f;
  for (int v = t; v < V; v += 256) sum += expf(row[v] - rowmax);
  red[t] = sum; __syncthreads();
  for (int o = 128; o > 0; o >>= 1) {
    if (t < o) red[t] += red[t + o];
    __syncthreads();
  }
  if (t == 0) {
    ws[STATS_OFF + bid * 2]     = rowmax;
    ws[STATS_OFF + bid * 2 + 1] = red[0];
  }
}

// -------- logits -> attention weights, in place (one expf per logit) ----
__global__ void k_attn(float* __restrict__ ws) {
  size_t idx = (size_t)blockIdx.x * 256 + threadIdx.x;   // over 16*NBTOT
  int lvl, row;
  if (idx < (size_t)S * NB0) {
    lvl = 0; row = (int)(idx >> 12);                     // /NB0
  } else if (idx < (size_t)S * (NB0 + NB1)) {
    lvl = 1; row = (int)((idx - (size_t)S * NB0) >> 15); // /NB1
  } else {
    lvl = 2; row = (int)((idx - (size_t)S * (NB0 + NB1)) >> 18); // /NB2
  }
  float mx  = ws[STATS_OFF + (lvl * 16 + row) * 2];
  float inv = 1.f / ws[STATS_OFF + (lvl * 16 + row) * 2 + 1];
  float x = ws[LOGITS_OFF + idx];
  ws[LOGITS_OFF + idx] = expf(x - mx) * inv;   // -inf (empty bin) -> 0
}

// -------- out = attn @ mean via WMMA; K (=V) split 32 ways, atomic acc --
__global__ void k_out(const float* __restrict__ ws, float* __restrict__ out) {
  int wid = blockIdx.x * 8 + (threadIdx.x >> 5);   // 768 waves
  int lane = threadIdx.x & 31;
  int lvl = wid >> 8;
  int rem = wid & 255;
  int ctile = rem >> 5;      // which 16 channels of C=128
  int kpart = rem & 31;      // which V slice
  int binoff, V, d; lvl_params(lvl, binoff, V, d);
  int Kchunk = V >> 5;
  int vstart = kpart * Kchunk;

  int row = lane & 15;
  int khi = (lane >> 4) << 1;
  int cch = ctile * 16 + (lane & 15);

  const float* ap = ws + LOGITS_OFF + (size_t)S * binoff + (size_t)row * V
                    + vstart + khi;                       // attn row, K-slice
  const float* bp = ws + SUMS_OFF
                    + ((size_t)binoff + vstart + khi) * C + cch;  // mean col

  v8f acc = {};
  #pragma unroll 4
  for (int it = 0; it < Kchunk / 4; ++it) {
    v2f a = *(const v2f*)ap;            // attn[row][kk], attn[row][kk+1]
    v2f b;
    b.x = bp[0];                        // mean[kk  ][cch]
    b.y = bp[C];                        // mean[kk+1][cch]
    acc = __builtin_amdgcn_wmma_f32_16x16x4_f32(false, a, false, b,
                                                (short)0, acc, false, false);
    ap += 4; bp += 4 * C;
  }

  int mbase = (lane >> 4) << 3;
  #pragma unroll
  for (int r = 0; r < 8; ++r)
    atomicAdd(&out[(size_t)(lvl * 16 + r + mbase) * C + cch], acc[r]);
}

extern "C" void kernel_launch(void* const* d_in, const int* in_sizes, int n_in,
                              void* d_out, int out_size, void* d_ws, size_t ws_size,
                              hipStream_t stream) {
  const float* coords = (const float*)d_in[0];
  const float* tokens = (const float*)d_in[1];
  const float* q0     = (const float*)d_in[2];
  const float* q1     = (const float*)d_in[3];
  const float* q2     = (const float*)d_in[4];
  float* out = (float*)d_out;
  float* ws  = (float*)d_ws;

  int B  = out_size / (SROWS * C);                 // 4
  long long BN = (long long)in_sizes[1] / C;       // B*N
  int N  = (int)(BN / B);                          // 100000

  for (int b = 0; b < B; ++b) {
    const float* cb = coords + (size_t)b * N * 3;
    const float* tb = tokens + (size_t)b * N * C;
    float* ob = out + (size_t)b * SROWS * C;

    // zero minmax/counts/sums (logits fully overwritten)
    hipMemsetAsync(ws, 0, (size_t)LOGITS_OFF * sizeof(float), stream);
    k_init  <<<1, 256, 0, stream>>>((unsigned*)ws, ob);
    k_minmax<<<128, 256, 0, stream>>>(cb, N, (unsigned*)ws);
    k_bin   <<<(N + 7) / 8, 256, 0, stream>>>(cb, tb, N, (const unsigned*)ws, ws);
    k_mean  <<<(NBTOT * 32) / 256, 256, 0, stream>>>(ws);
    k_logits<<<2336, 256, 0, stream>>>(q0, q1, q2, ws);   // 18688 waves, exact
    k_stats <<<48, 256, 0, stream>>>(ws);
    k_attn  <<<(S * NBTOT) / 256, 256, 0, stream>>>(ws);  // 18688 blocks, exact
    k_out   <<<96, 256, 0, stream>>>(ws, ob);             // 768 waves, exact
  }
}